// MultiBranchAutoregressiveDecoder_89352499626492
// MI455X (gfx1250) — compile-verified
//
#include <hip/hip_runtime.h>
#include <math.h>

// ---------------------------------------------------------------------------
// MultiBranchAutoregressiveDecoder on MI455X (gfx1250, wave32)
//
// KV-cached autoregressive decoding, one workgroup (128 thr = 4 waves) per
// batch element, all state resident in WGP LDS (~278 KB).  Prefill (n=288)
// uses V_WMMA_F32_16X16X4_F32 for every GEMM (D=16 -> 4 chained K=4 WMMAs)
// and for QK^T (HD=4 -> single WMMA per 16x16 tile) with flash-style online
// softmax.  The 11 incremental steps are tiny VALU work on cached K/V.
// ---------------------------------------------------------------------------

namespace {
constexpr int kBatch = 64, kTc = 288, kTf = 12, kD = 16, kHeads = 4, kHD = 4;
constexpr int kDF = 64, kNL = 4;
constexpr int kLMax = 304;                 // 300 padded to /16
constexpr int kNT = kTc / 16;              // 18 row tiles in prefill
constexpr float kNeg = -1e9f;
constexpr float kScale = 0.5f;             // 1/sqrt(HD)

// LDS layout (float offsets)
constexpr int O_XBUF = 0;
constexpr int O_QBUF = O_XBUF + kLMax * kD;
constexpr int O_ABUF = O_QBUF + kLMax * kD;
constexpr int O_KC   = O_ABUF + kLMax * kD;             // [NL][LMax][D]
constexpr int O_VC   = O_KC + kNL * kLMax * kD;
constexpr int O_WQ   = O_VC + kNL * kLMax * kD;         // [NL][48][16]
constexpr int O_WBQ  = O_WQ + kNL * 48 * 16;
constexpr int O_WO   = O_WBQ + kNL * 48;                // [NL][16][16]
constexpr int O_WBO  = O_WO + kNL * 256;
constexpr int O_WCB  = O_WBO + kNL * 16;
constexpr int O_G1   = O_WCB + kNL * 16;
constexpr int O_B1   = O_G1 + kNL * 16;
constexpr int O_G2   = O_B1 + kNL * 16;
constexpr int O_B2   = O_G2 + kNL * 16;
constexpr int O_G3   = O_B2 + kNL * 16;
constexpr int O_B3   = O_G3 + kNL * 16;
constexpr int O_W1   = O_B3 + kNL * 16;                 // [NL][64][16]
constexpr int O_WB1  = O_W1 + kNL * kDF * 16;
constexpr int O_W2   = O_WB1 + kNL * kDF;               // [NL][16][64]
constexpr int O_WB2  = O_W2 + kNL * 16 * kDF;
constexpr int O_OW   = O_WB2 + kNL * 16;                // out_W (16)
constexpr int O_OB   = O_OW + 16;                       // out_b
constexpr int O_HSCR = O_OB + 4;                        // 4 waves * 16*64
constexpr int O_TOK  = O_HSCR + 4 * 16 * kDF;           // 12 next-tokens
constexpr int O_TV   = O_TOK + 16;                      // current token act
constexpr int O_SA   = O_TV + 16;
constexpr int O_SB   = O_SA + 16;
constexpr int O_AS   = O_SB + 16;                       // attn head outputs
constexpr int O_HS   = O_AS + 16;                       // ffn hidden (64)
constexpr int O_QKVS = O_HS + kDF;                      // qkv vec (48)
constexpr int kSmemFloats = O_QKVS + 48;                // ~71k floats ~278KB
}  // namespace

typedef float v2f __attribute__((ext_vector_type(2)));
typedef float v8f __attribute__((ext_vector_type(8)));

__device__ __forceinline__ v8f vzero8() {
  v8f c = {0.f, 0.f, 0.f, 0.f, 0.f, 0.f, 0.f, 0.f};
  return c;
}

// D = A(16x4,f32) * B(4x16,f32) + C(16x16,f32) — the exact-fit CDNA5 op.
__device__ __forceinline__ v8f wmma4(v2f a, v2f b, v8f c) {
#if __has_builtin(__builtin_amdgcn_wmma_f32_16x16x4_f32)
  return __builtin_amdgcn_wmma_f32_16x16x4_f32(
      /*neg_a=*/false, a, /*neg_b=*/false, b,
      /*c_mod=*/(short)0, c, /*reuse_a=*/false, /*reuse_b=*/false);
#else
  // VALU fallback via lane shuffles (histogram would show wmma==0 if taken).
  int lane = threadIdx.x & 31;
  int n = lane & 15;
#pragma unroll
  for (int r = 0; r < 8; ++r) {
    int m = r + ((lane >> 4) << 3);
    float a0 = __shfl(a.x, m, 32),      a1 = __shfl(a.y, m, 32);
    float a2 = __shfl(a.x, m + 16, 32), a3 = __shfl(a.y, m + 16, 32);
    float q0 = __shfl(b.x, n, 32),      q1 = __shfl(b.y, n, 32);
    float q2 = __shfl(b.x, n + 16, 32), q3 = __shfl(b.y, n + 16, 32);
    c[r] += a0 * q0 + a1 * q1 + a2 * q2 + a3 * q3;
  }
  return c;
#endif
}

// Load WMMA A/B operand from row-major [row][k] storage (ld = row stride).
// A layout 16x4 f32: lanes 0-15 row=lane, VGPR0=K0/VGPR1=K1; lanes 16-31
// VGPR0=K2/VGPR1=K3.  B(4x16)=W^T loads with identical addressing on W[n][k].
__device__ __forceinline__ v2f ldAB(const float* base, int ld, int row0, int k0) {
  int lane = threadIdx.x & 31;
  const float* p = base + (row0 + (lane & 15)) * ld + k0 + ((lane >> 4) << 1);
  v2f v;
  v.x = p[0];
  v.y = p[1];
  return v;
}

__device__ __forceinline__ float rmax16(float v) {
#pragma unroll
  for (int o = 8; o; o >>= 1) v = fmaxf(v, __shfl_xor(v, o, 16));
  return v;
}
__device__ __forceinline__ float rsum16(float v) {
#pragma unroll
  for (int o = 8; o; o >>= 1) v += __shfl_xor(v, o, 16);
  return v;
}
__device__ __forceinline__ float rmax32(float v) {
#pragma unroll
  for (int o = 16; o; o >>= 1) v = fmaxf(v, __shfl_xor(v, o, 32));
  return v;
}
__device__ __forceinline__ float rsum32(float v) {
#pragma unroll
  for (int o = 16; o; o >>= 1) v += __shfl_xor(v, o, 32);
  return v;
}

// LayerNorm of a 16-vector in LDS, returning element n scaled by g,b.
__device__ __forceinline__ float ln16(const float* buf, int n, float g, float b) {
  float mu = 0.f;
#pragma unroll
  for (int k = 0; k < 16; ++k) mu += buf[k];
  mu *= (1.f / 16.f);
  float var = 0.f;
#pragma unroll
  for (int k = 0; k < 16; ++k) {
    float d = buf[k] - mu;
    var += d * d;
  }
  var *= (1.f / 16.f);
  return (buf[n] - mu) * rsqrtf(var + 1e-5f) * g + b;
}

extern "C" __global__ void __launch_bounds__(128, 1)
mbad_kernel(const float* __restrict__ b1, const float* __restrict__ b2,
            const float* __restrict__ b3, const float* __restrict__ fW,
            const float* __restrict__ fb, const float* __restrict__ pe,
            const float* __restrict__ Wqkv, const float* __restrict__ bqkv,
            const float* __restrict__ Wo, const float* __restrict__ bo,
            const float* __restrict__ cbo, const float* __restrict__ g1,
            const float* __restrict__ be1, const float* __restrict__ g2,
            const float* __restrict__ be2, const float* __restrict__ g3,
            const float* __restrict__ be3, const float* __restrict__ fW1,
            const float* __restrict__ fb1, const float* __restrict__ fW2,
            const float* __restrict__ fb2, const float* __restrict__ outW,
            const float* __restrict__ outB, float* __restrict__ out) {
  extern __shared__ float sm[];
  const int bidx = blockIdx.x;
  const int tid = threadIdx.x;
  const int lane = tid & 31;
  // force wave id scalar so task selection / loop bounds / Q-K-V destination
  // select compile to SALU branches instead of per-lane EXEC masking
  const int wave = __builtin_amdgcn_readfirstlane(tid >> 5);
  const int nthr = blockDim.x;      // 128
  const int nwav = nthr >> 5;       // 4

  // hint the input streams into cache (global_prefetch_b8)
  if (tid < 64) {
    __builtin_prefetch(b1 + (size_t)bidx * kTc * kD + tid * 64, 0, 3);
    __builtin_prefetch(b2 + (size_t)bidx * kTc * kD + tid * 64, 0, 3);
    __builtin_prefetch(b3 + (size_t)bidx * kTc * kD + tid * 64, 0, 3);
  }

  // ---- stage all weights into LDS ----
  auto cpy = [&](int dst, const float* src, int n) {
    for (int i = tid; i < n; i += nthr) sm[dst + i] = src[i];
  };
  cpy(O_WQ, Wqkv, kNL * 48 * 16);
  cpy(O_WBQ, bqkv, kNL * 48);
  cpy(O_WO, Wo, kNL * 256);
  cpy(O_WBO, bo, kNL * 16);
  cpy(O_WCB, cbo, kNL * 16);
  cpy(O_G1, g1, kNL * 16);  cpy(O_B1, be1, kNL * 16);
  cpy(O_G2, g2, kNL * 16);  cpy(O_B2, be2, kNL * 16);
  cpy(O_G3, g3, kNL * 16);  cpy(O_B3, be3, kNL * 16);
  cpy(O_W1, fW1, kNL * kDF * 16);
  cpy(O_WB1, fb1, kNL * kDF);
  cpy(O_W2, fW2, kNL * 16 * kDF);
  cpy(O_WB2, fb2, kNL * 16);
  cpy(O_OW, outW, 16);
  if (tid == 0) sm[O_OB] = outB[0];

  // ---- x0 = concat(b1,b2,b3) @ fusion_W^T + fusion_b + pos_emb ----
  {
    const float* pb0 = b1 + (size_t)bidx * kTc * kD;
    const float* pb1 = b2 + (size_t)bidx * kTc * kD;
    const float* pb2 = b3 + (size_t)bidx * kTc * kD;
    for (int idx = tid; idx < kTc * kD; idx += nthr) {
      int t = idx >> 4, d = idx & 15;
      float acc = fb[d];
      const float* w = fW + d * 48;
      const float* r0 = pb0 + t * kD;
      const float* r1 = pb1 + t * kD;
      const float* r2 = pb2 + t * kD;
#pragma unroll
      for (int j = 0; j < 16; ++j) acc += r0[j] * w[j];
#pragma unroll
      for (int j = 0; j < 16; ++j) acc += r1[j] * w[16 + j];
#pragma unroll
      for (int j = 0; j < 16; ++j) acc += r2[j] * w[32 + j];
      sm[O_XBUF + idx] = acc + pe[idx];
    }
  }
  __syncthreads();

  float* XB = sm + O_XBUF;
  float* QB = sm + O_QBUF;
  float* AB = sm + O_ABUF;

  // =========================== PREFILL (n = 288) ===========================
  for (int l = 0; l < kNL; ++l) {
    float* KCl = sm + O_KC + l * kLMax * kD;
    float* VCl = sm + O_VC + l * kLMax * kD;
    const float* wq   = sm + O_WQ  + l * 48 * 16;
    const float* wbq  = sm + O_WBQ + l * 48;
    const float* wo_l = sm + O_WO  + l * 256;
    const float* wbo  = sm + O_WBO + l * 16;
    const float* wcb  = sm + O_WCB + l * 16;
    const float* G1l  = sm + O_G1 + l * 16;
    const float* B1l  = sm + O_B1 + l * 16;
    const float* G2l  = sm + O_G2 + l * 16;
    const float* B2l  = sm + O_B2 + l * 16;
    const float* G3l  = sm + O_G3 + l * 16;
    const float* B3l  = sm + O_B3 + l * 16;
    const float* w1l  = sm + O_W1  + l * kDF * 16;
    const float* wb1l = sm + O_WB1 + l * kDF;
    const float* w2l  = sm + O_W2  + l * 16 * kDF;
    const float* wb2l = sm + O_WB2 + l * 16;

    // ---- P1: QKV projection, x(288x16) @ Wqkv^T(16x48), WMMA K=4 chain ----
    // One row tile per task: A fragments loaded once, reused for Q, K, V.
    for (int rt = wave; rt < kNT; rt += nwav) {
      v2f ax[4];
#pragma unroll
      for (int kc = 0; kc < 4; ++kc) ax[kc] = ldAB(XB, kD, rt * 16, kc * 4);
      int n = lane & 15;
      int rbase = rt * 16 + ((lane >> 4) << 3);
#pragma unroll
      for (int oc = 0; oc < 3; ++oc) {          // 0:Q 1:K 2:V (uniform)
        v8f c = vzero8();
#pragma unroll
        for (int kc = 0; kc < 4; ++kc)
          c = wmma4(ax[kc], ldAB(wq, kD, oc * 16, kc * 4), c);
        float bias = wbq[oc * 16 + n];
        float* dst = (oc == 0) ? QB : (oc == 1) ? KCl : VCl;  // scalar select
#pragma unroll
        for (int r = 0; r < 8; ++r)
          dst[(rbase + r) * kD + n] = c[r] + bias;
      }
    }
    __syncthreads();

    // ---- P2: causal flash attention per (head, query tile) ----
    for (int task = wave; task < kHeads * kNT; task += nwav) {
      int h = task / kNT, qt = task % kNT;
      int n = lane & 15;
      int rbase = qt * 16 + ((lane >> 4) << 3);
      float mi[8], li[8], op[8][4];
#pragma unroll
      for (int r = 0; r < 8; ++r) {
        mi[r] = -3e38f;
        li[r] = 0.f;
        op[r][0] = op[r][1] = op[r][2] = op[r][3] = 0.f;
      }
      v2f a = ldAB(QB + h * kHD, kD, qt * 16, 0);  // Q tile 16x4 (one head)
      for (int kt = 0; kt <= qt; ++kt) {
        v2f bm = ldAB(KCl + h * kHD, kD, kt * 16, 0);  // K^T tile 4x16
        v8f s = vzero8();
        s = wmma4(a, bm, s);                            // S 16x16, one WMMA
        int key = kt * 16 + n;
        const float* vr = VCl + key * kD + h * kHD;
        float vd0 = vr[0], vd1 = vr[1], vd2 = vr[2], vd3 = vr[3];
#pragma unroll
        for (int r = 0; r < 8; ++r) {
          int row = rbase + r;
          float sv = s[r] * kScale + (key > row ? kNeg : 0.f);
          float mnew = fmaxf(mi[r], rmax16(sv));
          float corr = __expf(mi[r] - mnew);
          float p = __expf(sv - mnew);
          li[r] = li[r] * corr + rsum16(p);
          mi[r] = mnew;
          op[r][0] = op[r][0] * corr + p * vd0;
          op[r][1] = op[r][1] * corr + p * vd1;
          op[r][2] = op[r][2] * corr + p * vd2;
          op[r][3] = op[r][3] * corr + p * vd3;
        }
      }
#pragma unroll
      for (int r = 0; r < 8; ++r) {
        float inv = 1.f / li[r];
#pragma unroll
        for (int d = 0; d < 4; ++d) {
          float o = rsum16(op[r][d]);
          if (n == 0) AB[(rbase + r) * kD + h * kHD + d] = o * inv;
        }
      }
    }
    __syncthreads();

    // ---- P3: out-proj + residual + LN1, then cross-bias + LN2 (fused) ----
    for (int rt = wave; rt < kNT; rt += nwav) {
      v8f c = vzero8();
#pragma unroll
      for (int kc = 0; kc < 4; ++kc) {
        v2f a = ldAB(AB, kD, rt * 16, kc * 4);
        v2f bm = ldAB(wo_l, kD, 0, kc * 4);
        c = wmma4(a, bm, c);
      }
      int n = lane & 15;
      int rbase = rt * 16 + ((lane >> 4) << 3);
      float gg1 = G1l[n], bb1 = B1l[n], cc = wcb[n];
      float gg2 = G2l[n], bb2 = B2l[n], boN = wbo[n];
#pragma unroll
      for (int r = 0; r < 8; ++r) {
        int row = rbase + r;
        float v = XB[row * kD + n] + c[r] + boN;
        float mu = rsum16(v) * (1.f / 16.f);
        float dv = v - mu;
        float var = rsum16(dv * dv) * (1.f / 16.f);
        float y = dv * rsqrtf(var + 1e-5f) * gg1 + bb1;
        float u = y + cc;                        // cross-attn bias only
        float mu2 = rsum16(u) * (1.f / 16.f);
        float du = u - mu2;
        float var2 = rsum16(du * du) * (1.f / 16.f);
        XB[row * kD + n] = du * rsqrtf(var2 + 1e-5f) * gg2 + bb2;
      }
    }
    __syncthreads();

    // ---- P4: FFN (16 -> 64 relu -> 16) + residual + LN3, per row tile ----
    {
      float* hs = sm + O_HSCR + wave * 16 * kDF;  // wave-private 16x64 tile
      for (int rt = wave; rt < kNT; rt += nwav) {
        int n = lane & 15;
        int rloc = ((lane >> 4) << 3);
        v2f ax[4];
#pragma unroll
        for (int kc = 0; kc < 4; ++kc) ax[kc] = ldAB(XB, kD, rt * 16, kc * 4);
#pragma unroll
        for (int nt = 0; nt < 4; ++nt) {
          v8f c = vzero8();
#pragma unroll
          for (int kc = 0; kc < 4; ++kc)
            c = wmma4(ax[kc], ldAB(w1l, kD, nt * 16, kc * 4), c);
          float bias = wb1l[nt * 16 + n];
#pragma unroll
          for (int r = 0; r < 8; ++r) {
            float v = c[r] + bias;
            hs[(rloc + r) * kDF + nt * 16 + n] = v > 0.f ? v : 0.f;
          }
        }
        // wave-private LDS tile: wait our own DS ops, no block barrier needed
        asm volatile("s_wait_dscnt 0" ::: "memory");
        v8f c2 = vzero8();
        for (int kc = 0; kc < 16; ++kc) {
          v2f a = ldAB(hs, kDF, 0, kc * 4);
          v2f bm = ldAB(w2l, kDF, 0, kc * 4);
          c2 = wmma4(a, bm, c2);
        }
        float gg3 = G3l[n], bb3 = B3l[n], b2n = wb2l[n];
        int rbase = rt * 16 + rloc;
#pragma unroll
        for (int r = 0; r < 8; ++r) {
          int row = rbase + r;
          float v = XB[row * kD + n] + c2[r] + b2n;
          float mu = rsum16(v) * (1.f / 16.f);
          float dv = v - mu;
          float var = rsum16(dv * dv) * (1.f / 16.f);
          XB[row * kD + n] = dv * rsqrtf(var + 1e-5f) * gg3 + bb3;
        }
      }
    }
    __syncthreads();
  }

  // next_token for step 0 (last prefill position 287)
  if (tid == 0) {
    float acc = sm[O_OB];
#pragma unroll
    for (int d = 0; d < kD; ++d) acc += XB[(kTc - 1) * kD + d] * sm[O_OW + d];
    sm[O_TOK + 0] = acc;
  }
  __syncthreads();

  // ====================== INCREMENTAL DECODE (11 steps) ====================
  for (int st = 1; st < kTf; ++st) {
    int p = kTc - 1 + st;  // position processed this step (288..298)
    if (tid < kD)
      sm[O_TV + tid] = sm[O_TOK + st - 1] * sm[O_OW + tid] + pe[p * kD + tid];
    __syncthreads();

    for (int l = 0; l < kNL; ++l) {
      float* KCl = sm + O_KC + l * kLMax * kD;
      float* VCl = sm + O_VC + l * kLMax * kD;
      const float* wq   = sm + O_WQ  + l * 48 * 16;
      const float* w1l  = sm + O_W1  + l * kDF * 16;
      const float* w2l  = sm + O_W2  + l * 16 * kDF;

      // qkv projection of the single token
      if (tid < 48) {
        float acc = sm[O_WBQ + l * 48 + tid];
        const float* wr = wq + tid * 16;
#pragma unroll
        for (int d = 0; d < 16; ++d) acc += sm[O_TV + d] * wr[d];
        sm[O_QKVS + tid] = acc;
      }
      __syncthreads();
      if (tid < kD) {  // append to KV cache
        KCl[p * kD + tid] = sm[O_QKVS + 16 + tid];
        VCl[p * kD + tid] = sm[O_QKVS + 32 + tid];
      }
      __syncthreads();

      // attention over p+1 cached keys: one wave per head
      {
        int h = wave;
        float q0 = sm[O_QKVS + h * 4 + 0], q1 = sm[O_QKVS + h * 4 + 1];
        float q2 = sm[O_QKVS + h * 4 + 2], q3 = sm[O_QKVS + h * 4 + 3];
        int kn = p + 1;
        float sarr[10];
        float mloc = -3e38f;
        int cnt = 0;
        for (int j = lane; j < kn; j += 32) {
          const float* kr = KCl + j * kD + h * kHD;
          float s = kScale * (q0 * kr[0] + q1 * kr[1] + q2 * kr[2] + q3 * kr[3]);
          sarr[cnt++] = s;
          mloc = fmaxf(mloc, s);
        }
        float m = rmax32(mloc);
        float lsum = 0.f, o0 = 0.f, o1 = 0.f, o2 = 0.f, o3 = 0.f;
        cnt = 0;
        for (int j = lane; j < kn; j += 32) {
          float pw = __expf(sarr[cnt++] - m);
          lsum += pw;
          const float* vr = VCl + j * kD + h * kHD;
          o0 += pw * vr[0]; o1 += pw * vr[1]; o2 += pw * vr[2]; o3 += pw * vr[3];
        }
        lsum = rsum32(lsum);
        o0 = rsum32(o0); o1 = rsum32(o1); o2 = rsum32(o2); o3 = rsum32(o3);
        if (lane == 0) {
          float inv = 1.f / lsum;
          sm[O_AS + h * 4 + 0] = o0 * inv;
          sm[O_AS + h * 4 + 1] = o1 * inv;
          sm[O_AS + h * 4 + 2] = o2 * inv;
          sm[O_AS + h * 4 + 3] = o3 * inv;
        }
      }
      __syncthreads();

      // out-proj + residual
      if (tid < kD) {
        float acc = sm[O_WBO + l * 16 + tid];
        const float* wr = sm + O_WO + l * 256 + tid * 16;
#pragma unroll
        for (int k = 0; k < 16; ++k) acc += sm[O_AS + k] * wr[k];
        sm[O_SA + tid] = sm[O_TV + tid] + acc;
      }
      __syncthreads();
      // LN1 then + cross-bias
      if (tid < kD) {
        float y = ln16(sm + O_SA, tid, sm[O_G1 + l * 16 + tid], sm[O_B1 + l * 16 + tid]);
        sm[O_SB + tid] = y + sm[O_WCB + l * 16 + tid];
      }
      __syncthreads();
      // LN2 -> x2
      if (tid < kD)
        sm[O_SA + tid] = ln16(sm + O_SB, tid, sm[O_G2 + l * 16 + tid], sm[O_B2 + l * 16 + tid]);
      __syncthreads();
      // FFN hidden
      if (tid < kDF) {
        float acc = sm[O_WB1 + l * kDF + tid];
        const float* wr = w1l + tid * 16;
#pragma unroll
        for (int d = 0; d < 16; ++d) acc += sm[O_SA + d] * wr[d];
        sm[O_HS + tid] = acc > 0.f ? acc : 0.f;
      }
      __syncthreads();
      // FFN out + residual
      if (tid < kD) {
        float acc = sm[O_WB2 + l * 16 + tid];
        const float* wr = w2l + tid * kDF;
#pragma unroll
        for (int k = 0; k < kDF; ++k) acc += sm[O_HS + k] * wr[k];
        sm[O_SB + tid] = sm[O_SA + tid] + acc;
      }
      __syncthreads();
      // LN3 -> new token activation
      if (tid < kD)
        sm[O_TV + tid] = ln16(sm + O_SB, tid, sm[O_G3 + l * 16 + tid], sm[O_B3 + l * 16 + tid]);
      __syncthreads();
    }

    if (tid == 0) {
      float acc = sm[O_OB];
#pragma unroll
      for (int d = 0; d < kD; ++d) acc += sm[O_TV + d] * sm[O_OW + d];
      sm[O_TOK + st] = acc;
    }
    __syncthreads();
  }

  // out[b][t] = dec[288+t] . out_W + out_b, with dec[288+t] = tok_t * out_W
  if (tid < kTf) {
    float s2 = 0.f;
#pragma unroll
    for (int d = 0; d < kD; ++d) s2 += sm[O_OW + d] * sm[O_OW + d];
    out[bidx * kTf + tid] = sm[O_TOK + tid] * s2 + sm[O_OB];
  }
}

extern "C" void kernel_launch(void* const* d_in, const int* in_sizes, int n_in,
                              void* d_out, int out_size, void* d_ws, size_t ws_size,
                              hipStream_t stream) {
  (void)in_sizes; (void)n_in; (void)out_size; (void)d_ws; (void)ws_size;
  const float* b1   = (const float*)d_in[0];
  const float* b2   = (const float*)d_in[1];
  const float* b3   = (const float*)d_in[2];
  const float* fW   = (const float*)d_in[3];
  const float* fb   = (const float*)d_in[4];
  const float* pe   = (const float*)d_in[5];
  const float* Wqkv = (const float*)d_in[6];
  const float* bqkv = (const float*)d_in[7];
  const float* Wo   = (const float*)d_in[8];
  const float* bo   = (const float*)d_in[9];
  const float* cbo  = (const float*)d_in[10];
  const float* g1   = (const float*)d_in[11];
  const float* be1  = (const float*)d_in[12];
  const float* g2   = (const float*)d_in[13];
  const float* be2  = (const float*)d_in[14];
  const float* g3   = (const float*)d_in[15];
  const float* be3  = (const float*)d_in[16];
  const float* fW1  = (const float*)d_in[17];
  const float* fb1  = (const float*)d_in[18];
  const float* fW2  = (const float*)d_in[19];
  const float* fb2  = (const float*)d_in[20];
  const float* outW = (const float*)d_in[21];
  const float* outB = (const float*)d_in[22];
  float* out = (float*)d_out;

  const size_t smemBytes = (size_t)kSmemFloats * sizeof(float);
  // CDNA5 WGP has 320KB LDS; opt in to large dynamic LDS (deterministic,
  // not a stream op, safe under graph capture).
  (void)hipFuncSetAttribute(reinterpret_cast<const void*>(mbad_kernel),
                            hipFuncAttributeMaxDynamicSharedMemorySize,
                            (int)smemBytes);
  mbad_kernel<<<dim3(kBatch), dim3(128), smemBytes, stream>>>(
      b1, b2, b3, fW, fb, pe, Wqkv, bqkv, Wo, bo, cbo, g1, be1, g2, be2, g3,
      be3, fW1, fb1, fW2, fb2, outW, outB, out);
}